// Model_80513456931018
// MI455X (gfx1250) — compile-verified
//
#include <hip/hip_runtime.h>
#include <hip/hip_bf16.h>

typedef __attribute__((ext_vector_type(2))) float v2f;
typedef __attribute__((ext_vector_type(8))) float v8f;

#define HIDDEN 128

static inline int cdiv(long long a, long long b) { return (int)((a + b - 1) / b); }

// ---------------------------------------------------------------------------
// deg init: deg[n] = 1.0 (self-loop weight)
__global__ void k_init_deg(float* __restrict__ deg, int n) {
    int i = blockIdx.x * blockDim.x + threadIdx.x;
    if (i < n) deg[i] = 1.0f;
}

// deg scatter: deg[dst[e]] += ew[e]
__global__ void k_deg_scatter(const int* __restrict__ dst, const float* __restrict__ ew,
                              float* __restrict__ deg, int nE) {
    int e = blockIdx.x * blockDim.x + threadIdx.x;
    if (e < nE) atomicAdd(&deg[dst[e]], ew[e]);
}

// dinv[n] = rsqrt(deg[n])  (deg >= 1 always due to self loop)
__global__ void k_rsqrt(float* __restrict__ dinv, int n) {
    int i = blockIdx.x * blockDim.x + threadIdx.x;
    if (i < n) dinv[i] = __frsqrt_rn(dinv[i]);
}

// zero a float buffer
__global__ void k_zero(float* __restrict__ p, long long n) {
    long long i = (long long)blockIdx.x * blockDim.x + threadIdx.x;
    if (i < n) p[i] = 0.0f;
}

// ---------------------------------------------------------------------------
// Layer-1 GEMM via V_WMMA_F32_16X16X4_F32:  h[N,128] = x[N,4] @ W1[4,128]
// FULL TILES ONLY (nTiles = N/16): no bounds checks, no divergence.
// One wave32 computes a 16-row x 128-col tile with 8 WMMAs (A reused).
// A 16x4 f32 layout: lane L<16 holds {x[M=L,K=0], x[M=L,K=1]},
//                    lane L>=16 holds {x[M=L-16,K=2], x[M=L-16,K=3]}
// B 4x16 f32 layout: lane L<16 holds {W1[K=0,N=L], W1[K=1,N=L]},
//                    lane L>=16 holds {W1[K=2,N=L-16], W1[K=3,N=L-16]}
// D 16x16 f32: VGPR i -> M = (lane>=16 ? 8 : 0)+i, N = lane&15
__global__ void k_gemm1_wmma(const float* __restrict__ x, const float* __restrict__ W1,
                             float* __restrict__ h, int nTiles) {
    int wave = (blockIdx.x * blockDim.x + threadIdx.x) >> 5;
    int lane = threadIdx.x & 31;
    if (wave >= nTiles) return;
    int row0 = wave * 16;

    int m     = lane & 15;
    int khalf = (lane >> 4) << 1;   // 0 or 2
    int arow  = row0 + m;

    v2f a;
    a.x = x[arow * 4 + khalf];
    a.y = x[arow * 4 + khalf + 1];

    int dm0  = (lane >> 4) << 3;    // 0 or 8
    int dcol = lane & 15;

    // one base pointer each; loads/stores lower to immediate-offset b32 ops
    const float* wbase = W1 + khalf * HIDDEN + m;
    float*       dbase = h + (long long)(row0 + dm0) * HIDDEN + dcol;

#pragma unroll
    for (int t = 0; t < 8; ++t) {
        v2f b;
        b.x = wbase[t * 16];
        b.y = wbase[t * 16 + HIDDEN];
        v8f c = {};
        c = __builtin_amdgcn_wmma_f32_16x16x4_f32(
            false, a, false, b, (short)0, c, false, false);
#pragma unroll
        for (int i = 0; i < 8; ++i)
            dbase[t * 16 + i * HIDDEN] = c[i];
    }
}

// Scalar tail for the last N%16 rows (launched only when N%16 != 0).
// One thread per (row, col): h[row,col] = sum_k x[row,k]*W1[k,col]
__global__ void k_gemm1_tail(const float* __restrict__ x, const float* __restrict__ W1,
                             float* __restrict__ h, int rowStart, int n) {
    int t   = blockIdx.x * blockDim.x + threadIdx.x;
    int row = rowStart + (t >> 7);
    int col = t & (HIDDEN - 1);
    if (row >= n) return;
    float4 xv = *(const float4*)(x + row * 4);
    float v = xv.x * W1[0 * HIDDEN + col] + xv.y * W1[1 * HIDDEN + col] +
              xv.z * W1[2 * HIDDEN + col] + xv.w * W1[3 * HIDDEN + col];
    h[(long long)row * HIDDEN + col] = v;
}

// ---------------------------------------------------------------------------
// Layer-1 edge scatter: wave per edge, lane owns a channel-quad.
// acc[dst] += dinv[src]*ew*dinv[dst] * h[src]   (4 atomic f32 adds / lane)
__global__ void k_scatter1(const int* __restrict__ src, const int* __restrict__ dst,
                           const float* __restrict__ ew, const float* __restrict__ dinv,
                           const float* __restrict__ h, float* __restrict__ acc, int nE) {
    long long t = (long long)blockIdx.x * blockDim.x + threadIdx.x;
    int e    = (int)(t >> 5);
    int lane = (int)(t & 31);
    if (e >= nE) return;
    int s = src[e], d = dst[e];
    float norm = dinv[s] * ew[e] * dinv[d];
    float4 hv = ((const float4*)(h + (long long)s * HIDDEN))[lane];
    float* ap = acc + (long long)d * HIDDEN + lane * 4;
    atomicAdd(ap + 0, norm * hv.x);
    atomicAdd(ap + 1, norm * hv.y);
    atomicAdd(ap + 2, norm * hv.z);
    atomicAdd(ap + 3, norm * hv.w);
}

// Layer-1 finalize (in place): acc = relu(acc + dinv[n]^2 * h + b1)
__global__ void k_final1(float* __restrict__ acc, const float* __restrict__ h,
                         const float* __restrict__ dinv, const float* __restrict__ b1,
                         int n) {
    long long i = (long long)blockIdx.x * blockDim.x + threadIdx.x;
    if (i >= (long long)n * HIDDEN) return;
    int node = (int)(i >> 7);
    int c    = (int)(i & (HIDDEN - 1));
    float di = dinv[node];
    float v  = acc[i] + di * di * h[i] + b1[c];
    acc[i] = v > 0.0f ? v : 0.0f;
}

// ---------------------------------------------------------------------------
// Layer-2 GEMV: wave per node, 128-dot via float4 + shuffle reduction.
// Also writes the self-loop + bias init into out.
__global__ void k_gemv2(const float* __restrict__ z, const float* __restrict__ W2,
                        const float* __restrict__ dinv, const float* __restrict__ b2,
                        float* __restrict__ h2, float* __restrict__ out, int n) {
    int wave = (blockIdx.x * blockDim.x + threadIdx.x) >> 5;
    int lane = threadIdx.x & 31;
    if (wave >= n) return;
    float4 zv = ((const float4*)(z + (long long)wave * HIDDEN))[lane];
    float4 wv = ((const float4*)W2)[lane];
    float s = zv.x * wv.x + zv.y * wv.y + zv.z * wv.z + zv.w * wv.w;
#pragma unroll
    for (int off = 16; off > 0; off >>= 1) s += __shfl_down(s, off, 32);
    if (lane == 0) {
        h2[wave] = s;
        float di = dinv[wave];
        out[wave] = di * di * s + b2[0];
    }
}

// Layer-2 edge scatter: out[dst] += dinv[src]*ew*dinv[dst] * h2[src]
__global__ void k_scatter2(const int* __restrict__ src, const int* __restrict__ dst,
                           const float* __restrict__ ew, const float* __restrict__ dinv,
                           const float* __restrict__ h2, float* __restrict__ out, int nE) {
    int e = blockIdx.x * blockDim.x + threadIdx.x;
    if (e >= nE) return;
    int s = src[e], d = dst[e];
    float norm = dinv[s] * ew[e] * dinv[d];
    atomicAdd(&out[d], norm * h2[s]);
}

// ---------------------------------------------------------------------------
extern "C" void kernel_launch(void* const* d_in, const int* in_sizes, int n_in,
                              void* d_out, int out_size, void* d_ws, size_t ws_size,
                              hipStream_t stream) {
    const float* x   = (const float*)d_in[0];   // [N,4]
    const int*   ei  = (const int*)d_in[1];     // [2,E]
    const float* ew  = (const float*)d_in[2];   // [E]
    const float* W1  = (const float*)d_in[3];   // [4,128]
    const float* b1  = (const float*)d_in[4];   // [128]
    const float* W2  = (const float*)d_in[5];   // [128,1]
    const float* b2  = (const float*)d_in[6];   // [1]
    float* out = (float*)d_out;                 // [N,1]

    const int N = in_sizes[0] / 4;
    const int E = in_sizes[2];
    const int* src = ei;
    const int* dst = ei + E;

    // workspace layout (512B aligned slabs)
    char* ws = (char*)d_ws;
    auto align512 = [](size_t v) { return (v + 511) & ~(size_t)511; };
    size_t off = 0;
    float* dinv = (float*)(ws + off); off += align512((size_t)N * sizeof(float));
    float* h    = (float*)(ws + off); off += align512((size_t)N * HIDDEN * sizeof(float));
    float* acc  = (float*)(ws + off); off += align512((size_t)N * HIDDEN * sizeof(float));
    float* h2   = (float*)(ws + off); off += align512((size_t)N * sizeof(float));
    (void)ws_size;

    const int B = 256;

    // degree / normalization
    k_init_deg<<<cdiv(N, B), B, 0, stream>>>(dinv, N);
    k_deg_scatter<<<cdiv(E, B), B, 0, stream>>>(dst, ew, dinv, E);
    k_rsqrt<<<cdiv(N, B), B, 0, stream>>>(dinv, N);

    // zero layer-1 accumulator
    k_zero<<<cdiv((long long)N * HIDDEN, B), B, 0, stream>>>(acc, (long long)N * HIDDEN);

    // layer-1 GEMM: WMMA f32 16x16x4 over full 16-row tiles, scalar tail
    {
        int nTiles = N / 16;                    // full tiles only
        if (nTiles > 0) {
            long long threads = (long long)nTiles * 32;
            k_gemm1_wmma<<<cdiv(threads, B), B, 0, stream>>>(x, W1, h, nTiles);
        }
        int tail = N - nTiles * 16;             // host-side, deterministic in N
        if (tail > 0) {
            long long threads = (long long)tail * HIDDEN;
            k_gemm1_tail<<<cdiv(threads, B), B, 0, stream>>>(x, W1, h, nTiles * 16, N);
        }
    }

    // layer-1 aggregation + finalize (relu, bias, self loop)
    k_scatter1<<<cdiv((long long)E * 32, B), B, 0, stream>>>(src, dst, ew, dinv, h, acc, E);
    k_final1<<<cdiv((long long)N * HIDDEN, B), B, 0, stream>>>(acc, h, dinv, b1, N);

    // layer-2 GEMV + out init (self loop + bias), then scalar edge scatter
    k_gemv2<<<cdiv((long long)N * 32, B), B, 0, stream>>>(acc, W2, dinv, b2, h2, out, N);
    k_scatter2<<<cdiv(E, B), B, 0, stream>>>(src, dst, ew, dinv, h2, out, E);
}